// VectorQuantizerEMA_78297253806627
// MI455X (gfx1250) — compile-verified
//
#include <hip/hip_runtime.h>
#include <hip/hip_bf16.h>

// ---------------------------------------------------------------------------
// VQ-VAE vector quantizer for MI455X (gfx1250, wave32).
//   inputs:      [16,64,64,256] f32  -> N=65536 rows, D=256
//   embeddings:  [256,1024]     f32  -> K=1024 codes
// Fused GEMM(x·e) + argmax(x·e - 0.5||e||^2) via V_WMMA_F32_16X16X4_F32.
// N-blocked by 4 (one A-frag feeds 4 WMMAs) with XOR-swizzled LDS so both
// staging writes and compute reads are bank-conflict-free and the fully
// unrolled K loop uses immediate-offset ds_load_b64.
// Outputs (concat, f32):
//   quantized [16777216] | commitment_loss [1] | encoding_indices [65536]
// ---------------------------------------------------------------------------

typedef __attribute__((ext_vector_type(2))) float v2f;
typedef __attribute__((ext_vector_type(8))) float v8f;

#define VQ_D     256
#define VQ_K     1024
#define VQ_N     65536
#define VQ_BETA  0.25f

// LDS layout (bytes):
//   [0      .. 65536)  B block: 4 column tiles, tile t4 at t4*16384
//   [65536  .. 196608) A tiles: wave w at 65536 + w*16384
//   [196608 .. 200704) h staged (1024 f32)
//   [200704 .. 201216) winning indices (128 int)
#define SMEM_B_OFF   0
#define SMEM_A_OFF   65536
#define SMEM_H_OFF   196608
#define SMEM_IDX_OFF 200704
#define SMEM_TOTAL   201216

// XOR bank swizzle: logical (q, n) -> physical float2 index within a tile.
// Conflict-free for both stride-1-in-q staging writes and the WMMA fragment
// reads (permutation inside each 128B block; half-waves 32 banks apart).
#define SWZ(q, n) (((q) * 16) + ((n) ^ (((q) >> 1) & 15)))

// ---------------------------------------------------------------------------
// Prep: h[k] = 0.5*||e_k||^2, Et[k][d] = E[d][k] (transposed codebook so the
// gather epilogue is coalesced), zero the loss accumulator (deterministic
// across graph replays).
// ---------------------------------------------------------------------------
__global__ void vq_prep(const float* __restrict__ E,
                        float* __restrict__ h,
                        float* __restrict__ Et,
                        float* __restrict__ loss) {
  const int k = blockIdx.x * blockDim.x + threadIdx.x;
  if (k == 0) *loss = 0.0f;
  if (k >= VQ_K) return;
  float s = 0.0f;
  for (int d = 0; d < VQ_D; ++d) {
    const float v = E[d * VQ_K + k];     // coalesced across k
    Et[k * VQ_D + d] = v;
    s += v * v;
  }
  h[k] = 0.5f * s;
}

// ---------------------------------------------------------------------------
// Main fused kernel: 256 threads = 8 waves, 128 rows per block, 512 blocks.
// ---------------------------------------------------------------------------
__launch_bounds__(256)
__global__ void vq_main(const float* __restrict__ X,
                        const float* __restrict__ Et,
                        const float* __restrict__ h,
                        float* __restrict__ q_out,
                        float* __restrict__ loss,
                        float* __restrict__ idx_out) {
  extern __shared__ char smem[];
  float2* Bb   = (float2*)(smem + SMEM_B_OFF);   // 4 tiles, 2048 float2 each
  float*  hs   = (float*) (smem + SMEM_H_OFF);
  int*    idxs = (int*)   (smem + SMEM_IDX_OFF);

  const int t    = threadIdx.x;
  const int wave = t >> 5;
  const int lane = t & 31;
  const int jj   = lane >> 4;       // half-wave: K-pair {0,1} vs {2,3}
  const int nl   = lane & 15;       // column / row-in-half index
  const int row0 = blockIdx.x * 128;

  float2* Aw = (float2*)(smem + SMEM_A_OFF + wave * 16384);

  // Stage h once.
  for (int i = t; i < VQ_K; i += 256) hs[i] = h[i];

  // Stage this wave's 16x256 A tile into swizzled WMMA A-fragment layout:
  //   logical (q = 2d+j, m); lane<16 consumes K{4d,4d+1}, lane>=16 K{4d+2,4d+3}.
  {
    const float2* src = (const float2*)(X + (size_t)(row0 + wave * 16) * VQ_D);
    for (int it = 0; it < 64; ++it) {
      const int p  = it * 32 + lane;     // coalesced global b64 reads
      const int m  = p >> 7;             // row 0..15
      const int qq = p & 127;            // float2 pair index within row
      Aw[SWZ(qq, m)] = src[m * 128 + qq];
    }
  }

  float bestv[8];
  int   besti[8];
#pragma unroll
  for (int r = 0; r < 8; ++r) { bestv[r] = -3.4e38f; besti[r] = 0; }

  // 16 steps x 64 codes (4 column tiles of 16) per step.
  for (int step = 0; step < 16; ++step) {
    const int n0 = step * 64;
    __syncthreads();                     // also covers hs/A on step 0
    // Stage 256x64 B block from Et (coalesced global, conflict-free LDS).
#pragma unroll
    for (int it = 0; it < 32; ++it) {
      const int p  = it * 256 + t;       // 0..8191
      const int qq = p & 127;
      const int ng = p >> 7;             // code within block: 0..63
      Bb[(ng >> 4) * 2048 + SWZ(qq, ng & 15)] =
          ((const float2*)(Et + (size_t)(n0 + ng) * VQ_D))[qq];
    }
    __syncthreads();

    v8f c0 = {0.f,0.f,0.f,0.f,0.f,0.f,0.f,0.f};
    v8f c1 = c0, c2 = c0, c3 = c0;
#pragma unroll
    for (int d2 = 0; d2 < 128; d2 += 2) {
      // d2 even => ((d2+jj)>>1)&15 == (d2>>1)&15 : compile-time swizzle mask.
      const int col = (d2 + jj) * 16 + (nl ^ ((d2 >> 1) & 15));
      const v2f a  = *(const v2f*)&Aw[col];
      const v2f b0 = *(const v2f*)&Bb[col];
      const v2f b1 = *(const v2f*)&Bb[2048 + col];
      const v2f b2 = *(const v2f*)&Bb[4096 + col];
      const v2f b3 = *(const v2f*)&Bb[6144 + col];
      c0 = __builtin_amdgcn_wmma_f32_16x16x4_f32(false, a, false, b0, (short)0, c0, false, false);
      c1 = __builtin_amdgcn_wmma_f32_16x16x4_f32(false, a, false, b1, (short)0, c1, false, false);
      c2 = __builtin_amdgcn_wmma_f32_16x16x4_f32(false, a, false, b2, (short)0, c2, false, false);
      c3 = __builtin_amdgcn_wmma_f32_16x16x4_f32(false, a, false, b3, (short)0, c3, false, false);
    }

    // score = x·e - 0.5||e||^2 ; argmin(distance) == argmax(score).
    // Tiles visited in ascending code order; strict '>' keeps first minimum.
    const float hn0 = hs[n0 + nl];
    const float hn1 = hs[n0 + 16 + nl];
    const float hn2 = hs[n0 + 32 + nl];
    const float hn3 = hs[n0 + 48 + nl];
#pragma unroll
    for (int r = 0; r < 8; ++r) {
      float s;
      s = c0[r] - hn0; if (s > bestv[r]) { bestv[r] = s; besti[r] = n0 + nl; }
      s = c1[r] - hn1; if (s > bestv[r]) { bestv[r] = s; besti[r] = n0 + 16 + nl; }
      s = c2[r] - hn2; if (s > bestv[r]) { bestv[r] = s; besti[r] = n0 + 32 + nl; }
      s = c3[r] - hn3; if (s > bestv[r]) { bestv[r] = s; besti[r] = n0 + 48 + nl; }
    }
  }

  // Cross-lane argmax inside each 16-lane half (C layout: lanes0-15 rows0-7,
  // lanes16-31 rows8-15, N = lane&15). Tie-break: smaller index wins.
#pragma unroll
  for (int r = 0; r < 8; ++r) {
    float v = bestv[r];
    int   i = besti[r];
#pragma unroll
    for (int m = 8; m >= 1; m >>= 1) {
      const float ov = __shfl_xor(v, m, 32);
      const int   oi = __shfl_xor(i, m, 32);
      if (ov > v || (ov == v && oi < i)) { v = ov; i = oi; }
    }
    besti[r] = i;
  }
  if (nl == 0) {
#pragma unroll
    for (int r = 0; r < 8; ++r) idxs[wave * 16 + jj * 8 + r] = besti[r];
  }
  __syncthreads();

  // Epilogue: gather winning code rows (coalesced from Et), write
  // quantized = x + (q - x) (straight-through expression), accumulate MSE.
  float acc = 0.0f;
  const int myrow0 = row0 + wave * 16;
  for (int m = 0; m < 16; ++m) {
    const int k = idxs[wave * 16 + m];
    const float4* cv = (const float4*)(Et + (size_t)k * VQ_D);
    const float4* xv = (const float4*)(X + (size_t)(myrow0 + m) * VQ_D);
    float4*       ov = (float4*)(q_out + (size_t)(myrow0 + m) * VQ_D);
#pragma unroll
    for (int it = 0; it < 2; ++it) {
      const int e = lane + it * 32;      // 64 float4 per row
      const float4 q4 = cv[e];
      const float4 x4 = xv[e];
      const float dx = q4.x - x4.x, dy = q4.y - x4.y;
      const float dz = q4.z - x4.z, dw = q4.w - x4.w;
      float4 o;
      o.x = x4.x + dx; o.y = x4.y + dy; o.z = x4.z + dz; o.w = x4.w + dw;
      ov[e] = o;
      acc += dx * dx + dy * dy + dz * dz + dw * dw;
    }
  }
#pragma unroll
  for (int m = 16; m >= 1; m >>= 1) acc += __shfl_xor(acc, m, 32);
  if (lane == 0)
    atomicAdd(loss, acc * (VQ_BETA / (float)(VQ_N * VQ_D)));

  // Indices output (reference dtype lowered to f32 by the harness).
  if (t < 128) idx_out[row0 + t] = (float)idxs[t];
}

// ---------------------------------------------------------------------------
extern "C" void kernel_launch(void* const* d_in, const int* in_sizes, int n_in,
                              void* d_out, int out_size, void* d_ws, size_t ws_size,
                              hipStream_t stream) {
  (void)in_sizes; (void)n_in; (void)out_size; (void)ws_size;
  const float* X = (const float*)d_in[0];   // inputs  [65536,256]
  const float* E = (const float*)d_in[1];   // codebook [256,1024]

  float* out  = (float*)d_out;
  float* q    = out;                         // [16777216]
  float* loss = out + (size_t)VQ_N * VQ_D;   // [1]
  float* idx  = loss + 1;                    // [65536]

  float* h  = (float*)d_ws;                  // [1024]
  float* Et = h + VQ_K;                      // [1024*256]

  vq_prep<<<4, 256, 0, stream>>>(E, h, Et, loss);
  vq_main<<<512, 256, SMEM_TOTAL, stream>>>(X, Et, h, q, loss, idx);
}